// DogDetector_25769803776070
// MI455X (gfx1250) — compile-verified
//
#include <hip/hip_runtime.h>
#include <math.h>

// ---------------- problem constants (match reference) ----------------
#define B_IMG    16
#define N_ANCH   200000
#define PRE_K    1024
#define MAX_DET  100
#define CONF_THR 0.3f
#define NMS_THR  0.5f
#define MIN_BOX  0.01f

// ---------------- tuning ----------------
#define NBINS     8192        // 13-bit bins of monotone score key
#define BIN_SHIFT 19
#define CHUNKS    32          // blocks per image for streaming passes
#define CHUNK     ((N_ANCH + CHUNKS - 1) / CHUNKS)
#define CAP       2048        // candidate capacity per image (>= PRE_K + bin population)

// monotone float->uint mapping: larger float => larger key
__device__ __forceinline__ unsigned mono_key(float f) {
  unsigned u = __float_as_uint(f);
  return (u & 0x80000000u) ? ~u : (u | 0x80000000u);
}
__device__ __forceinline__ float unmono(unsigned k) {
  unsigned u = (k & 0x80000000u) ? (k ^ 0x80000000u) : ~k;
  return __uint_as_float(u);
}
__device__ __forceinline__ float clamp01(float x) { return fminf(fmaxf(x, 0.0f), 1.0f); }

// async global->LDS path (gfx1250), guarded so compile never breaks
#if defined(__HIP_DEVICE_COMPILE__) && defined(__gfx1250__)
#if __has_builtin(__builtin_amdgcn_global_load_async_to_lds_b64) && \
    __has_builtin(__builtin_amdgcn_s_wait_asynccnt)
#define USE_ASYNC_LDS 1
#endif
#endif

#ifdef USE_ASYNC_LDS
typedef int v2i __attribute__((ext_vector_type(2)));
typedef __attribute__((address_space(1))) v2i* g_v2i_p;   // global (AS1)
typedef __attribute__((address_space(3))) v2i* l_v2i_p;   // LDS (AS3)
#endif

// ---------------- kernel 1: zero hist + counters ----------------
__global__ void k_zero(unsigned* __restrict__ p, int n) {
  int i = blockIdx.x * blockDim.x + threadIdx.x;
  if (i < n) p[i] = 0u;
}

// ---------------- kernel 2: per-image histogram of score keys ----------------
__global__ void k_hist(const float* __restrict__ conf, unsigned* __restrict__ hist) {
  __shared__ unsigned lh[NBINS];
  for (int i = threadIdx.x; i < NBINS; i += blockDim.x) lh[i] = 0u;
  __syncthreads();

  int img   = blockIdx.x / CHUNKS;
  int chunk = blockIdx.x - img * CHUNKS;
  int begin = chunk * CHUNK;
  int end   = begin + CHUNK; if (end > N_ANCH) end = N_ANCH;
  const float* c = conf + (size_t)img * N_ANCH;

  for (int n = begin + threadIdx.x; n < end; n += blockDim.x) {
    __builtin_prefetch(c + n + 2048, 0, 0);  // gfx1250 global_prefetch_b8
    unsigned key = mono_key(c[n]);
    atomicAdd(&lh[key >> BIN_SHIFT], 1u);
  }
  __syncthreads();

  unsigned* gh = hist + (size_t)img * NBINS;
  for (int i = threadIdx.x; i < NBINS; i += blockDim.x) {
    unsigned v = lh[i];
    if (v) atomicAdd(&gh[i], v);
  }
}

// ---------------- kernel 3: find threshold bin per image ----------------
__global__ void k_select(const unsigned* __restrict__ hist, unsigned* __restrict__ bstar) {
  int b = threadIdx.x;
  if (b >= B_IMG) return;
  const unsigned* h = hist + (size_t)b * NBINS;
  unsigned acc = 0;
  int bs = 0;
  for (int bin = NBINS - 1; bin >= 0; --bin) {
    acc += h[bin];
    if (acc >= PRE_K) { bs = bin; break; }
  }
  bstar[b] = (unsigned)bs;
}

// ---------------- kernel 4: collect candidates in bins >= threshold ----------------
__global__ void k_collect(const float* __restrict__ conf,
                          const unsigned* __restrict__ bstar,
                          unsigned* __restrict__ cnt,
                          unsigned long long* __restrict__ cand) {
  int img   = blockIdx.x / CHUNKS;
  int chunk = blockIdx.x - img * CHUNKS;
  unsigned bs = bstar[img];
  int begin = chunk * CHUNK;
  int end   = begin + CHUNK; if (end > N_ANCH) end = N_ANCH;
  const float* c = conf + (size_t)img * N_ANCH;
  unsigned long long* cd = cand + (size_t)img * CAP;

  for (int n = begin + threadIdx.x; n < end; n += blockDim.x) {
    __builtin_prefetch(c + n + 2048, 0, 0);
    unsigned key = mono_key(c[n]);
    if ((key >> BIN_SHIFT) >= bs) {
      unsigned pos = atomicAdd(&cnt[img], 1u);
      if (pos < CAP) {
        // tie-break: equal score -> lower index ranks higher (matches lax.top_k)
        cd[pos] = ((unsigned long long)key << 32) | (unsigned)(~(unsigned)n);
      }
    }
  }
}

// ---------------- kernel 5: sort + decode + greedy NMS + top-100 emit ----------------
__global__ __launch_bounds__(PRE_K) void k_nms(const float* __restrict__ bbox,
                                               const float* __restrict__ anchors,
                                               const unsigned long long* __restrict__ cand,
                                               const unsigned* __restrict__ cnt,
                                               float* __restrict__ out) {
  __shared__ unsigned long long s_combo[CAP];   // 16 KB
  __shared__ float s_box[PRE_K][4];             // 16 KB
  __shared__ int   s_keep[PRE_K];               // 4 KB
  __shared__ int   s_rank[PRE_K];               // 4 KB

  const int b = blockIdx.x;
  const int t = threadIdx.x;
  const unsigned long long* g = cand + (size_t)b * CAP;
  int M = (int)cnt[b]; if (M > CAP) M = CAP;

#ifdef USE_ASYNC_LDS
  for (int i = t; i < CAP; i += PRE_K) {
    __builtin_amdgcn_global_load_async_to_lds_b64(
        (g_v2i_p)(g + i), (l_v2i_p)&s_combo[i], 0, 0);
  }
  __builtin_amdgcn_s_wait_asynccnt(0);
  __syncthreads();
  for (int i = t; i < CAP; i += PRE_K) if (i >= M) s_combo[i] = 0ull;
#else
  for (int i = t; i < CAP; i += PRE_K) s_combo[i] = (i < M) ? g[i] : 0ull;
#endif
  __syncthreads();

  // bitonic sort, descending (largest combo first)
  for (int kk = 2; kk <= CAP; kk <<= 1) {
    for (int j = kk >> 1; j > 0; j >>= 1) {
      for (int i = t; i < CAP; i += PRE_K) {
        int ixj = i ^ j;
        if (ixj > i) {
          bool desc = ((i & kk) == 0);
          unsigned long long a = s_combo[i], c2 = s_combo[ixj];
          if (desc ? (a < c2) : (a > c2)) { s_combo[i] = c2; s_combo[ixj] = a; }
        }
      }
      __syncthreads();
    }
  }

  // decode my (rank-t) candidate
  unsigned long long cb = s_combo[t];
  unsigned key = (unsigned)(cb >> 32);
  unsigned idx = ~(unsigned)(cb & 0xFFFFFFFFull);
  float myscore = -1.0f;
  float bx0 = 0.f, by0 = 0.f, bx1 = 0.f, by1 = 0.f;
  if (idx < (unsigned)N_ANCH) {
    float cf = unmono(key);
    myscore = 1.0f / (1.0f + expf(-cf));
    const float* a = anchors + (size_t)idx * 4;
    float aw  = a[2] - a[0];
    float ah  = a[3] - a[1];
    float acx = a[0] + 0.5f * aw;
    float acy = a[1] + 0.5f * ah;
    const float* d = bbox + ((size_t)b * N_ANCH + idx) * 4;
    float dx = d[0], dy = d[1];
    float dw = fminf(fmaxf(d[2], -4.0f), 4.0f);
    float dh = fminf(fmaxf(d[3], -4.0f), 4.0f);
    float cx = dx * aw + acx;
    float cy = dy * ah + acy;
    float pw = expf(dw) * aw;
    float ph = expf(dh) * ah;
    bx0 = clamp01(cx - 0.5f * pw);
    by0 = clamp01(cy - 0.5f * ph);
    bx1 = clamp01(cx + 0.5f * pw);
    by1 = clamp01(cy + 0.5f * ph);
  }
  s_box[t][0] = bx0; s_box[t][1] = by0; s_box[t][2] = bx1; s_box[t][3] = by1;
  s_keep[t] = (idx < (unsigned)N_ANCH && myscore > CONF_THR) ? 1 : 0;
  __syncthreads();

  // greedy NMS: sequential over sorted rows, parallel over columns
  float myarea = (bx1 - bx0) * (by1 - by0);
  for (int i = 0; i < PRE_K - 1; ++i) {
    if (s_keep[i]) {
      float rx0 = s_box[i][0], ry0 = s_box[i][1], rx1 = s_box[i][2], ry1 = s_box[i][3];
      if (t > i && s_keep[t]) {
        float iw = fminf(rx1, bx1) - fmaxf(rx0, bx0);
        float ih = fminf(ry1, by1) - fmaxf(ry0, by0);
        iw = fmaxf(iw, 0.0f); ih = fmaxf(ih, 0.0f);
        float inter = iw * ih;
        float uni = (rx1 - rx0) * (ry1 - ry0) + myarea - inter;
        float iou = inter / fmaxf(uni, 1e-9f);
        if (iou > NMS_THR) s_keep[t] = 0;
      }
    }
    __syncthreads();
  }

  // inclusive scan of keep flags -> output slot for each survivor
  s_rank[t] = s_keep[t];
  __syncthreads();
  for (int off = 1; off < PRE_K; off <<= 1) {
    int v = (t >= off) ? s_rank[t - off] : 0;
    __syncthreads();
    s_rank[t] += v;
    __syncthreads();
  }
  int rank = s_rank[t] - s_keep[t];   // exclusive rank among kept

  // outputs: fb [B,100,4] | fs [B,100] | ok [B,100] (floats 0/1)
  float* fb = out;
  float* fs = out + (size_t)B_IMG * MAX_DET * 4;
  float* fo = fs + (size_t)B_IMG * MAX_DET;
  if (t < MAX_DET) {
    size_t s = (size_t)b * MAX_DET + t;
    fb[s * 4 + 0] = 0.f; fb[s * 4 + 1] = 0.f; fb[s * 4 + 2] = 0.f; fb[s * 4 + 3] = 0.f;
    fs[s] = 0.f; fo[s] = 0.f;
  }
  __syncthreads();
  if (s_keep[t] && rank < MAX_DET) {
    bool ok = (myscore > CONF_THR) && ((bx1 - bx0) >= MIN_BOX) && ((by1 - by0) >= MIN_BOX);
    if (ok) {
      size_t s = (size_t)b * MAX_DET + rank;
      fb[s * 4 + 0] = bx0; fb[s * 4 + 1] = by0; fb[s * 4 + 2] = bx1; fb[s * 4 + 3] = by1;
      fs[s] = myscore;
      fo[s] = 1.0f;
    }
  }
}

// ---------------- host launcher ----------------
extern "C" void kernel_launch(void* const* d_in, const int* in_sizes, int n_in,
                              void* d_out, int out_size, void* d_ws, size_t ws_size,
                              hipStream_t stream) {
  const float* bbox    = (const float*)d_in[0];  // [B,N,4]
  const float* conf    = (const float*)d_in[1];  // [B,N]
  const float* anchors = (const float*)d_in[2];  // [N,4]
  float* out = (float*)d_out;

  // workspace layout
  unsigned* hist  = (unsigned*)d_ws;                         // B*NBINS
  unsigned* cnt   = hist + (size_t)B_IMG * NBINS;            // B
  unsigned* bstar = cnt + B_IMG;                             // B
  unsigned long long* cand =
      (unsigned long long*)(void*)(bstar + B_IMG);           // B*CAP (8B aligned)

  int nz = B_IMG * NBINS + B_IMG;  // hist + cnt
  hipLaunchKernelGGL(k_zero,    dim3((nz + 255) / 256), dim3(256), 0, stream, hist, nz);
  hipLaunchKernelGGL(k_hist,    dim3(B_IMG * CHUNKS),   dim3(256), 0, stream, conf, hist);
  hipLaunchKernelGGL(k_select,  dim3(1),                dim3(32),  0, stream, hist, bstar);
  hipLaunchKernelGGL(k_collect, dim3(B_IMG * CHUNKS),   dim3(256), 0, stream,
                     conf, bstar, cnt, cand);
  hipLaunchKernelGGL(k_nms,     dim3(B_IMG),            dim3(PRE_K), 0, stream,
                     bbox, anchors, cand, cnt, out);
}